// SAGE_20710332301837
// MI455X (gfx1250) — compile-verified
//
#include <hip/hip_runtime.h>
#include <hip/hip_bf16.h>

// ---------------------------------------------------------------------------
// GraphSAGE 3-layer forward for MI455X (gfx1250, wave32).
// Roofline: edge scatter moves ~2 GB (640K edges x 512B gather + 512B atomic
// x 3 layers) -> ~100us floor @ 23.3 TB/s; GEMMs are only ~8.2 GFLOP, so we
// keep exact fp32 precision on the WMMA pipe (V_WMMA_F32_16X16X4_F32).
// Weights are staged transposed into LDS (once per block, ~130KB of the
// 320KB/WGP) so every B fragment is one ds_load_b64 instead of global loads.
// ---------------------------------------------------------------------------

typedef float v2f __attribute__((ext_vector_type(2)));
typedef float v8f __attribute__((ext_vector_type(8)));

#define N_NODES 50000
#define N_EDGES 640000
#define KDIM 128
#define BN_EPS 1e-5f
#define STK (KDIM + 2)   // padded K-stride for transposed LDS weights

// ---------------- degree / inverse degree ----------------------------------
__global__ void degree_kernel(const int* __restrict__ dst, float* __restrict__ deg,
                              int n_edges) {
    int e = blockIdx.x * blockDim.x + threadIdx.x;
    if (e < n_edges) unsafeAtomicAdd(&deg[dst[e]], 1.0f);
}

__global__ void invdeg_kernel(float* __restrict__ deg, int n) {
    int i = blockIdx.x * blockDim.x + threadIdx.x;
    if (i < n) deg[i] = 1.0f / fmaxf(deg[i], 1.0f);
}

// ---------------- edge scatter-add (mean aggregation numerator) ------------
// one wave32 per edge; lane l moves float4 at column 4*l (coalesced 512B row).
// src/dst indices are wave-uniform -> force scalar loads via readfirstlane.
__global__ void scatter_add(const float* __restrict__ h,
                            const int* __restrict__ src,
                            const int* __restrict__ dst,
                            float* __restrict__ agg, int n_edges) {
    int tid  = blockIdx.x * blockDim.x + threadIdx.x;
    int edge = tid >> 5;
    int lane = tid & 31;
    if (edge >= n_edges) return;          // uniform per wave
    edge = __builtin_amdgcn_readfirstlane(edge);
    const int s = src[edge];              // scalar load (uniform address)
    const int d = dst[edge];
    const float4 v = ((const float4*)(h + (size_t)s * KDIM))[lane];
    float* dp = agg + (size_t)d * KDIM + lane * 4;
    unsafeAtomicAdd(dp + 0, v.x);
    unsafeAtomicAdd(dp + 1, v.y);
    unsafeAtomicAdd(dp + 2, v.z);
    unsafeAtomicAdd(dp + 3, v.w);
}

// ---------------- fused SAGE conv: out = H@Ws + (agg*invdeg)@Wn + b --------
// 256 threads = 8 waves; block covers 128 rows x (NT*16) cols.
// Weights staged TRANSPOSED into dynamic LDS with padded stride STK so the
// (ka, ka+1) B-fragment pair is one aligned ds_load_b64 per matrix per tile.
// f32 WMMA fragment layouts per CDNA5 ISA 7.12.2:
//   A 16x4 : lane%16 = M row; lane/16 selects K pair {0,1} vs {2,3}
//   B 4x16 : lane%16 = N col; VGPR r holds K rows r / r+2
//   C/D    : VGPR i holds rows i (lanes 0-15) / i+8 (lanes 16-31)
template <int NT>
__global__ void sage_conv_wmma(const float* __restrict__ H,
                               const float* __restrict__ AGG,
                               const float* __restrict__ invdeg,
                               const float* __restrict__ Ws,   // K x N row-major
                               const float* __restrict__ Wn,   // K x N row-major
                               const float* __restrict__ bias, // N
                               float* __restrict__ out,        // rows x N
                               int n_rows) {
    constexpr int N = NT * 16;
    constexpr int NSHIFT = (NT == 8) ? 7 : 6;

    extern __shared__ float lds[];        // 2 * N * STK floats
    float* wS = lds;
    float* wN = lds + N * STK;

    // stage both weight matrices, transposed: wX[c*STK + k] = WX[k*N + c]
    for (int i = threadIdx.x; i < KDIM * N; i += 256) {
        const int r = i >> NSHIFT;        // k index
        const int c = i & (N - 1);        // column
        wS[c * STK + r] = Ws[i];
        wN[c * STK + r] = Wn[i];
    }
    __syncthreads();

    const int wave = threadIdx.x >> 5;
    const int lane = threadIdx.x & 31;
    const int half = lane >> 4;
    const int ml   = lane & 15;

    const int row0 = blockIdx.x * 128 + wave * 16;
    const int row  = row0 + ml;                   // A-fragment row for this lane
    const int rowc = (row < n_rows) ? row : 0;    // clamp: no EXEC divergence
    const float valid = (row < n_rows) ? 1.0f : 0.0f;
    const float idg   = invdeg[rowc] * valid;     // fold mean-normalize + mask

    const float* __restrict__ hrow = H   + (size_t)rowc * KDIM;
    const float* __restrict__ arow = AGG + (size_t)rowc * KDIM;
    __builtin_prefetch(hrow, 0, 1);               // global_prefetch_b8
    __builtin_prefetch(arow, 0, 1);

    int wb[NT];
#pragma unroll
    for (int t = 0; t < NT; ++t) wb[t] = (t * 16 + ml) * STK;

    v8f acc[NT] = {};

    for (int k = 0; k < KDIM; k += 4) {
        const int ka = k + 2 * half;
        v2f aH = *(const v2f*)(hrow + ka);        // global_load_b64
        v2f aA = *(const v2f*)(arow + ka);
        aH.x *= valid; aH.y *= valid;
        aA.x *= idg;   aA.y *= idg;
#pragma unroll
        for (int t = 0; t < NT; ++t) {
            const v2f bS = *(const v2f*)(wS + wb[t] + ka);  // ds_load_b64
            const v2f bN = *(const v2f*)(wN + wb[t] + ka);
            acc[t] = __builtin_amdgcn_wmma_f32_16x16x4_f32(
                false, aH, false, bS, (short)0, acc[t], false, false);
            acc[t] = __builtin_amdgcn_wmma_f32_16x16x4_f32(
                false, aA, false, bN, (short)0, acc[t], false, false);
        }
    }

#pragma unroll
    for (int t = 0; t < NT; ++t) {
        const int col = t * 16 + ml;
        const float bv = bias[col];
#pragma unroll
        for (int i = 0; i < 8; ++i) {
            const int r = row0 + i + 8 * half;
            if (r < n_rows) out[(size_t)r * N + col] = acc[t][i] + bv;
        }
    }
}

// ---------------- batchnorm: column stats + fused apply/ReLU ---------------
__global__ void bn_stats(const float* __restrict__ h, float* __restrict__ stats,
                         int n_rows) {
    const int c = threadIdx.x;                    // blockDim = 128, coalesced
    float s = 0.0f, ss = 0.0f;
    for (int r = blockIdx.x; r < n_rows; r += gridDim.x) {
        float v = h[(size_t)r * KDIM + c];
        s += v;
        ss += v * v;
    }
    unsafeAtomicAdd(&stats[c], s);
    unsafeAtomicAdd(&stats[KDIM + c], ss);
}

__global__ void bn_apply_relu(float* __restrict__ h,
                              const float* __restrict__ stats,
                              const float* __restrict__ gamma,
                              const float* __restrict__ beta, int n_rows) {
    const int idx = blockIdx.x * blockDim.x + threadIdx.x;
    if (idx >= n_rows * KDIM) return;
    const int c = idx & (KDIM - 1);
    const float inv_n = 1.0f / (float)n_rows;
    const float mu  = stats[c] * inv_n;
    const float var = stats[KDIM + c] * inv_n - mu * mu;
    const float v = (h[idx] - mu) * rsqrtf(var + BN_EPS) * gamma[c] + beta[c];
    h[idx] = fmaxf(v, 0.0f);
}

// ---------------------------------------------------------------------------
extern "C" void kernel_launch(void* const* d_in, const int* in_sizes, int n_in,
                              void* d_out, int out_size, void* d_ws, size_t ws_size,
                              hipStream_t stream) {
    const float* x   = (const float*)d_in[0];
    const int*   src = (const int*)d_in[1];
    const int*   dst = (const int*)d_in[2];
    const float* Ws0 = (const float*)d_in[3];
    const float* Wn0 = (const float*)d_in[4];
    const float* b0  = (const float*)d_in[5];
    const float* Ws1 = (const float*)d_in[6];
    const float* Wn1 = (const float*)d_in[7];
    const float* b1  = (const float*)d_in[8];
    const float* Ws2 = (const float*)d_in[9];
    const float* Wn2 = (const float*)d_in[10];
    const float* b2  = (const float*)d_in[11];
    const float* g0  = (const float*)d_in[12];
    const float* be0 = (const float*)d_in[13];
    const float* g1  = (const float*)d_in[14];
    const float* be1 = (const float*)d_in[15];
    float* out = (float*)d_out;

    // workspace layout (~77 MB total)
    const size_t featBytes = (size_t)N_NODES * KDIM * sizeof(float); // 25.6 MB
    char* ws = (char*)d_ws;
    float* deg   = (float*)(ws);                             // 200 KB (pad 204800)
    float* agg   = (float*)(ws + 204800);                    // featBytes
    float* h_a   = (float*)(ws + 204800 + featBytes);        // featBytes
    float* h_b   = (float*)(ws + 204800 + 2 * featBytes);    // featBytes
    float* stats = (float*)(ws + 204800 + 3 * featBytes);    // 1 KB

    const int convBlocks    = (N_NODES + 127) / 128;         // 391
    const int scatterBlocks = (N_EDGES * 32) / 256;          // 80000
    const int applyBlocks   = (N_NODES * KDIM) / 256;        // 25000
    const size_t shmem8 = (size_t)2 * 128 * STK * sizeof(float); // ~130 KB
    const size_t shmem4 = (size_t)2 * 64  * STK * sizeof(float); // ~65 KB

    // ---- degrees (shared by all layers) ----
    hipMemsetAsync(deg, 0, (size_t)N_NODES * sizeof(float), stream);
    degree_kernel<<<(N_EDGES + 255) / 256, 256, 0, stream>>>(dst, deg, N_EDGES);
    invdeg_kernel<<<(N_NODES + 255) / 256, 256, 0, stream>>>(deg, N_NODES);

    // ---- layer 0: x -> h_a (128 -> 128), BN+ReLU in place ----
    hipMemsetAsync(agg, 0, featBytes, stream);
    scatter_add<<<scatterBlocks, 256, 0, stream>>>(x, src, dst, agg, N_EDGES);
    sage_conv_wmma<8><<<convBlocks, 256, shmem8, stream>>>(x, agg, deg, Ws0, Wn0,
                                                           b0, h_a, N_NODES);
    hipMemsetAsync(stats, 0, 2 * KDIM * sizeof(float), stream);
    bn_stats<<<512, 128, 0, stream>>>(h_a, stats, N_NODES);
    bn_apply_relu<<<applyBlocks, 256, 0, stream>>>(h_a, stats, g0, be0, N_NODES);

    // ---- layer 1: h_a -> h_b (128 -> 128), BN+ReLU in place ----
    hipMemsetAsync(agg, 0, featBytes, stream);
    scatter_add<<<scatterBlocks, 256, 0, stream>>>(h_a, src, dst, agg, N_EDGES);
    sage_conv_wmma<8><<<convBlocks, 256, shmem8, stream>>>(h_a, agg, deg, Ws1, Wn1,
                                                           b1, h_b, N_NODES);
    hipMemsetAsync(stats, 0, 2 * KDIM * sizeof(float), stream);
    bn_stats<<<512, 128, 0, stream>>>(h_b, stats, N_NODES);
    bn_apply_relu<<<applyBlocks, 256, 0, stream>>>(h_b, stats, g1, be1, N_NODES);

    // ---- layer 2: h_b -> out (128 -> 64), no BN ----
    hipMemsetAsync(agg, 0, featBytes, stream);
    scatter_add<<<scatterBlocks, 256, 0, stream>>>(h_b, src, dst, agg, N_EDGES);
    sage_conv_wmma<4><<<convBlocks, 256, shmem4, stream>>>(h_b, agg, deg, Ws2, Wn2,
                                                           b2, out, N_NODES);
}